// Top2Gate_68934225101309
// MI455X (gfx1250) — compile-verified
//
#include <hip/hip_runtime.h>
#include <hip/hip_bf16.h>

#define S 8192
#define D 1024
#define E 64
#define CAP 256
static const size_t SEC = (size_t)S * E * CAP; // 134,217,728

typedef __attribute__((ext_vector_type(2))) float v2f;
typedef __attribute__((ext_vector_type(4))) float v4f;
typedef __attribute__((ext_vector_type(8))) float v8f;

// ---------------------------------------------------------------------------
// Kernel 1: logits = x @ wg^T via V_WMMA_F32_16X16X4_F32.
// Block = 128 threads (4 waves). Each block owns a 16-token row tile; each
// wave owns one 16-expert column tile (4 * 16 = E). Four accumulators break
// the WMMA->WMMA accumulation dependency chain.
// A frag (16x4 f32): lanes 0-15 hold M=lane, K = {0,1}; lanes 16-31 K = {2,3}.
// B frag (4x16 f32): lanes 0-15 hold N=lane, K = {0,1}; lanes 16-31 K = {2,3}.
// ---------------------------------------------------------------------------
__global__ __launch_bounds__(128) void moe_gate_gemm(const float* __restrict__ x,
                                                     const float* __restrict__ wg,
                                                     float* __restrict__ logits) {
  const int lane = threadIdx.x & 31;
  const int wave = threadIdx.x >> 5;
  const int row0 = blockIdx.x * 16;
  const int col0 = wave * 16;
  const int m    = lane & 15;
  const int koff = (lane >> 4) * 2;

  const float* ap = x  + (size_t)(row0 + m) * D + koff;
  const float* bp = wg + (size_t)(col0 + m) * D + koff;

  v8f acc0 = {}, acc1 = {}, acc2 = {}, acc3 = {};
  for (int k = 0; k < D; k += 16) {
    v2f a0 = *(const v2f*)(ap + k + 0);
    v2f b0 = *(const v2f*)(bp + k + 0);
    v2f a1 = *(const v2f*)(ap + k + 4);
    v2f b1 = *(const v2f*)(bp + k + 4);
    v2f a2 = *(const v2f*)(ap + k + 8);
    v2f b2 = *(const v2f*)(bp + k + 8);
    v2f a3 = *(const v2f*)(ap + k + 12);
    v2f b3 = *(const v2f*)(bp + k + 12);
    acc0 = __builtin_amdgcn_wmma_f32_16x16x4_f32(false, a0, false, b0, (short)0, acc0, false, false);
    acc1 = __builtin_amdgcn_wmma_f32_16x16x4_f32(false, a1, false, b1, (short)0, acc1, false, false);
    acc2 = __builtin_amdgcn_wmma_f32_16x16x4_f32(false, a2, false, b2, (short)0, acc2, false, false);
    acc3 = __builtin_amdgcn_wmma_f32_16x16x4_f32(false, a3, false, b3, (short)0, acc3, false, false);
  }
  v8f acc = (acc0 + acc1) + (acc2 + acc3);

  // C/D layout: VGPR v -> M = v + 8*(lane/16), N = lane&15
  const int mbase = (lane >> 4) * 8;
#pragma unroll
  for (int v = 0; v < 8; ++v) {
    logits[(size_t)(row0 + mbase + v) * E + (col0 + m)] = acc[v];
  }
}

// ---------------------------------------------------------------------------
// Kernel 2: per-token softmax over E=64, top-1 and top-2 (first-index ties,
// matching jnp.argmax), store full gates (needed for aux loss) + raw top gate
// probabilities.
// ---------------------------------------------------------------------------
__global__ __launch_bounds__(256) void moe_gate_softmax(const float* __restrict__ logits,
                                                        float* __restrict__ gates,
                                                        int* __restrict__ idx1,
                                                        int* __restrict__ idx2,
                                                        float* __restrict__ g1raw,
                                                        float* __restrict__ g2raw) {
  const int t = blockIdx.x * blockDim.x + threadIdx.x;
  if (t >= S) return;
  const float* lp = logits + (size_t)t * E;

  float l[E];
#pragma unroll
  for (int e = 0; e < E; e += 4) {
    v4f v = *(const v4f*)(lp + e);
    l[e + 0] = v.x; l[e + 1] = v.y; l[e + 2] = v.z; l[e + 3] = v.w;
  }

  float m1 = -__builtin_inff(); int i1 = 0;
  float m2 = -__builtin_inff(); int i2 = 0;
#pragma unroll
  for (int e = 0; e < E; ++e) {
    float v = l[e];
    if (v > m1) { m2 = m1; i2 = i1; m1 = v; i1 = e; }
    else if (v > m2) { m2 = v; i2 = e; }
  }

  float sum = 0.f;
#pragma unroll
  for (int e = 0; e < E; ++e) { l[e] = __expf(l[e] - m1); sum += l[e]; }
  const float inv = 1.0f / sum;

  float* gp = gates + (size_t)t * E;
#pragma unroll
  for (int e = 0; e < E; e += 4) {
    v4f v = { l[e + 0] * inv, l[e + 1] * inv, l[e + 2] * inv, l[e + 3] * inv };
    *(v4f*)(gp + e) = v;
  }

  idx1[t] = i1;
  idx2[t] = i2;
  g1raw[t] = inv;                     // exp(0)/sum
  g2raw[t] = __expf(m2 - m1) * inv;
}

// ---------------------------------------------------------------------------
// Kernel 3: per-expert serial rank over tokens (deterministic cumsum).
// One lane per expert (64 lanes, 2 waves). All lanes read the same broadcast
// idx values (int4 = 4 tokens per step). Exactly one lane writes each rank.
// ---------------------------------------------------------------------------
__global__ __launch_bounds__(64) void moe_gate_scan(const int* __restrict__ idx1,
                                                    const int* __restrict__ idx2,
                                                    int* __restrict__ rank1,
                                                    int* __restrict__ rank2,
                                                    int* __restrict__ counts1) {
  const int e = threadIdx.x;
  int c1 = 0, c2 = 0;
  for (int s = 0; s < S; s += 4) {
    const int4 a = *(const int4*)(idx1 + s);
    const int4 b = *(const int4*)(idx2 + s);
    if (a.x == e) rank1[s + 0] = c1++;
    if (a.y == e) rank1[s + 1] = c1++;
    if (a.z == e) rank1[s + 2] = c1++;
    if (a.w == e) rank1[s + 3] = c1++;
    if (b.x == e) rank2[s + 0] = c2++;
    if (b.y == e) rank2[s + 1] = c2++;
    if (b.z == e) rank2[s + 2] = c2++;
    if (b.w == e) rank2[s + 3] = c2++;
  }
  counts1[e] = c1;
}

// ---------------------------------------------------------------------------
// Kernel 4: zero-fill the ~1.07 GB output with B128 nontemporal stores.
// This is the bandwidth-dominant phase (~46 us at 23.3 TB/s).
// ---------------------------------------------------------------------------
__global__ __launch_bounds__(256) void moe_gate_zero(float* __restrict__ out, long long n4) {
  const long long i = (long long)blockIdx.x * blockDim.x + threadIdx.x;
  if (i < n4) {
    v4f z = {0.f, 0.f, 0.f, 0.f};
    __builtin_nontemporal_store(z, (v4f*)out + i);
  }
  if (i == 0) out[2 * SEC] = 0.f;  // final tail float (total = 1 + 2*SEC)
}

// ---------------------------------------------------------------------------
// Kernel 5: aux loss. Lane e sums gates column e (coalesced across lanes,
// fixed order -> deterministic). l_aux = E * sum_e(me[e]*ce[e]).
// ---------------------------------------------------------------------------
__global__ __launch_bounds__(64) void moe_gate_aux(const float* __restrict__ gates,
                                                   const int* __restrict__ counts1,
                                                   float* __restrict__ out) {
  __shared__ float part[E];
  const int e = threadIdx.x;
  float sum = 0.f;
#pragma unroll 8
  for (int s = 0; s < S; ++s) sum += gates[(size_t)s * E + e];
  const float me = sum * (1.0f / S);
  const float ce = (float)counts1[e] * (1.0f / S);
  part[e] = me * ce;
  __syncthreads();
  if (e == 0) {
    float tot = 0.f;
#pragma unroll
    for (int k = 0; k < E; ++k) tot += part[k];
    out[0] = tot * (float)E;  // mean over E * E^2 == E * sum
  }
}

// ---------------------------------------------------------------------------
// Kernel 6: scatter the (at most) 2 nonzero combine/dispatch entries per token.
// ---------------------------------------------------------------------------
__global__ __launch_bounds__(256) void moe_gate_scatter(const int* __restrict__ idx1,
                                                        const int* __restrict__ idx2,
                                                        const int* __restrict__ rank1,
                                                        const int* __restrict__ rank2,
                                                        const int* __restrict__ counts1,
                                                        const float* __restrict__ g1raw,
                                                        const float* __restrict__ g2raw,
                                                        float* __restrict__ out) {
  const int t = blockIdx.x * blockDim.x + threadIdx.x;
  if (t >= S) return;
  const int i1 = idx1[t];
  const int i2 = idx2[t];
  const int l1 = rank1[t];
  const int l2 = rank2[t] + counts1[i2];
  const bool keep1 = l1 < CAP;
  const bool keep2 = l2 < CAP;
  float g1 = keep1 ? g1raw[t] : 0.f;
  float g2 = keep2 ? g2raw[t] : 0.f;
  const float denom = fmaxf(g1 + g2, 1.1920928955078125e-07f);
  g1 /= denom;
  g2 /= denom;
  float* combine  = out + 1;
  float* dispatch = out + 1 + SEC;
  if (keep1) {
    const size_t off = ((size_t)t * E + i1) * CAP + (size_t)l1;
    combine[off] = g1;
    if (g1 != 0.f) dispatch[off] = 1.f;
  }
  if (keep2) {
    const size_t off = ((size_t)t * E + i2) * CAP + (size_t)l2;
    combine[off] = g2;
    if (g2 != 0.f) dispatch[off] = 1.f;
  }
}

extern "C" void kernel_launch(void* const* d_in, const int* in_sizes, int n_in,
                              void* d_out, int out_size, void* d_ws, size_t ws_size,
                              hipStream_t stream) {
  const float* x  = (const float*)d_in[0];  // [S, D]
  const float* wg = (const float*)d_in[1];  // [E, D]
  float* out = (float*)d_out;               // [1 + S*E*CAP + S*E*CAP]

  // Workspace layout (~4.4 MB)
  float* logits  = (float*)d_ws;            // S*E
  float* gates   = logits + (size_t)S * E;  // S*E
  int*   idx1    = (int*)(gates + (size_t)S * E);
  int*   idx2    = idx1 + S;
  int*   rank1   = idx2 + S;
  int*   rank2   = rank1 + S;
  int*   counts1 = rank2 + S;               // E
  float* g1raw   = (float*)(counts1 + E);
  float* g2raw   = g1raw + S;

  // 1) logits = x @ wg^T  (WMMA f32)
  moe_gate_gemm<<<dim3(S / 16), dim3(128), 0, stream>>>(x, wg, logits);

  // 2) softmax + top-2 per token
  moe_gate_softmax<<<dim3(S / 256), dim3(256), 0, stream>>>(logits, gates, idx1, idx2, g1raw, g2raw);

  // 3) deterministic per-expert token ranks
  moe_gate_scan<<<dim3(1), dim3(64), 0, stream>>>(idx1, idx2, rank1, rank2, counts1);

  // 4) zero-fill the full output (dominant cost, nontemporal B128 stores)
  const long long n4 = (long long)(2 * SEC) / 4;  // 67,108,864 float4's
  moe_gate_zero<<<dim3((unsigned)(n4 / 256)), dim3(256), 0, stream>>>(out, n4);

  // 5) aux loss -> out[0] (after zero-fill so the write sticks)
  moe_gate_aux<<<dim3(1), dim3(64), 0, stream>>>(gates, counts1, out);

  // 6) scatter nonzero combine/dispatch entries
  moe_gate_scatter<<<dim3(S / 256), dim3(256), 0, stream>>>(idx1, idx2, rank1, rank2,
                                                            counts1, g1raw, g2raw, out);
}